// Encoder_39402029974135
// MI455X (gfx1250) — compile-verified
//
#include <hip/hip_runtime.h>
#include <hip/hip_bf16.h>

typedef __attribute__((ext_vector_type(2))) float v2f;
typedef __attribute__((ext_vector_type(8))) float v8f;

#define NTHR 256

// ---- LDS layout (floats) ----
// sA: union region. Phase 1: x (9*512 @0), h0 (8*512 @4608), h1 (12*512 @8704).
//     Phase 2: xm buffer [t][c], stride 40, t=0..511  (20480 floats)
// sH2: h2 (16*512)
// weight pack offsets:
#define OW0  0      // 8*9*2   = 144
#define OB0  144    // 8
#define OG0  152    // 8   (pre-scaled gamma/sqrt(1+eps))
#define OBE0 160    // 8
#define OW1  168    // 12*8*2  = 192
#define OB1  360
#define OG1  372
#define OBE1 384
#define OW2  396    // 16*12*2 = 384
#define OB2  780
#define OG2  796
#define OBE2 812
#define OW3  828    // 32*16   = 512
#define OB3  1340   // 32
#define NWGT 1372

__device__ __forceinline__ float leaky(float x) {
    return x >= 0.f ? x : 0.01f * x;
}

__global__ __launch_bounds__(NTHR) void encoder_fused_kernel(
    const float* __restrict__ x,
    const float* __restrict__ w0, const float* __restrict__ b0,
    const float* __restrict__ g0, const float* __restrict__ be0,
    const float* __restrict__ w1, const float* __restrict__ b1,
    const float* __restrict__ g1, const float* __restrict__ be1,
    const float* __restrict__ w2, const float* __restrict__ b2,
    const float* __restrict__ g2, const float* __restrict__ be2,
    const float* __restrict__ w3, const float* __restrict__ b3,
    float* __restrict__ outD, float* __restrict__ outL)
{
    __shared__ float sA[20480];     // x/h0/h1 union, then xm [t*40+c]
    __shared__ float sH2[16 * 512];
    __shared__ float sEMP[32 * 33]; // emp -> shrunk -> L (stride 33)
    __shared__ float sCSCR[8 * 256];
    __shared__ float sWgt[NWGT];
    __shared__ float sRed[NTHR];
    __shared__ float sPS[NTHR];
    __shared__ float sMean[32];
    __shared__ float sScal[2];

    const int tid  = threadIdx.x;
    const int row  = blockIdx.x;          // 0..2047 == (b*64+s)
    float* sX  = sA;                      // 9*512
    float* sH0 = sA + 4608;               // 8*512
    float* sH1 = sA + 8704;               // 12*512
    float* sXM = sA;                      // reused: [t][c] stride 40

    // ---------------- load x row + weights into LDS ----------------
    {
        const float* xr = x + (size_t)row * 9 * 512;
        for (int i = tid; i < 9 * 512; i += NTHR) sX[i] = xr[i];
        const float bs = rsqrtf(1.0f + 1e-5f);
        for (int i = tid; i < 144; i += NTHR) sWgt[OW0 + i] = w0[i];
        for (int i = tid; i < 192; i += NTHR) sWgt[OW1 + i] = w1[i];
        for (int i = tid; i < 384; i += NTHR) sWgt[OW2 + i] = w2[i];
        for (int i = tid; i < 512; i += NTHR) sWgt[OW3 + i] = w3[i];
        if (tid < 8)  { sWgt[OB0 + tid] = b0[tid]; sWgt[OG0 + tid] = g0[tid] * bs; sWgt[OBE0 + tid] = be0[tid]; }
        if (tid < 12) { sWgt[OB1 + tid] = b1[tid]; sWgt[OG1 + tid] = g1[tid] * bs; sWgt[OBE1 + tid] = be1[tid]; }
        if (tid < 16) { sWgt[OB2 + tid] = b2[tid]; sWgt[OG2 + tid] = g2[tid] * bs; sWgt[OBE2 + tid] = be2[tid]; }
        if (tid < 32) { sWgt[OB3 + tid] = b3[tid]; }
    }
    __syncthreads();

    // ---------------- conv0: (9,512)->(8,511), BN, leaky ----------------
    for (int co = 0; co < 8; ++co) {
        float bb = sWgt[OB0 + co], gg = sWgt[OG0 + co], be = sWgt[OBE0 + co];
        for (int t = tid; t < 511; t += NTHR) {
            float acc = bb;
#pragma unroll
            for (int ci = 0; ci < 9; ++ci) {
                acc += sX[ci * 512 + t]     * sWgt[OW0 + (co * 9 + ci) * 2 + 0];
                acc += sX[ci * 512 + t + 1] * sWgt[OW0 + (co * 9 + ci) * 2 + 1];
            }
            sH0[co * 512 + t] = leaky(acc * gg + be);
        }
    }
    __syncthreads();

    // ---------------- conv1: (8,511)->(12,510) ----------------
    for (int co = 0; co < 12; ++co) {
        float bb = sWgt[OB1 + co], gg = sWgt[OG1 + co], be = sWgt[OBE1 + co];
        for (int t = tid; t < 510; t += NTHR) {
            float acc = bb;
#pragma unroll
            for (int ci = 0; ci < 8; ++ci) {
                acc += sH0[ci * 512 + t]     * sWgt[OW1 + (co * 8 + ci) * 2 + 0];
                acc += sH0[ci * 512 + t + 1] * sWgt[OW1 + (co * 8 + ci) * 2 + 1];
            }
            sH1[co * 512 + t] = leaky(acc * gg + be);
        }
    }
    __syncthreads();

    // ---------------- conv2: (12,510)->(16,509) ----------------
    for (int co = 0; co < 16; ++co) {
        float bb = sWgt[OB2 + co], gg = sWgt[OG2 + co], be = sWgt[OBE2 + co];
        for (int t = tid; t < 509; t += NTHR) {
            float acc = bb;
#pragma unroll
            for (int ci = 0; ci < 12; ++ci) {
                acc += sH1[ci * 512 + t]     * sWgt[OW2 + (co * 12 + ci) * 2 + 0];
                acc += sH1[ci * 512 + t + 1] * sWgt[OW2 + (co * 12 + ci) * 2 + 1];
            }
            sH2[co * 512 + t] = leaky(acc * gg + be);
        }
    }
    __syncthreads();

    // ---------------- conv3 (1x1, pad=1): -> xm [t][c], t=0..510 ----------------
    // x/h0/h1 are dead; sXM aliases them.
    {
        const int co = tid & 31, tg = tid >> 5;
        for (int t = tg; t < 511; t += 8) {
            float acc = sWgt[OB3 + co];
            if (t >= 1 && t <= 509) {
#pragma unroll
                for (int ci = 0; ci < 16; ++ci)
                    acc += sH2[ci * 512 + (t - 1)] * sWgt[OW3 + co * 16 + ci];
            }
            sXM[t * 40 + co] = acc;
        }
        if (tid < 32) sXM[511 * 40 + tid] = 0.f;   // K padding row
    }
    __syncthreads();

    // ---------------- column means, subtract in place ----------------
    {
        const int c = tid & 31, tg = tid >> 5;
        float ps = 0.f;
        for (int t = tg; t < 511; t += 8) ps += sXM[t * 40 + c];
        sPS[tid] = ps;
        __syncthreads();
        if (tid < 32) {
            float m = 0.f;
#pragma unroll
            for (int g = 0; g < 8; ++g) m += sPS[g * 32 + tid];
            sMean[tid] = m * (1.0f / 511.0f);
        }
        __syncthreads();
        float mc = sMean[c];
        for (int t = tg; t < 511; t += 8) sXM[t * 40 + c] -= mc;
    }
    __syncthreads();

    // ---------------- covariance: emp = xm^T xm / 510 via fp32 WMMA ----------------
    // 8 waves: 4 output tiles x 2 K-halves. K = 512 (padded), step 4.
    {
        const int wv   = tid >> 5;
        const int lane = tid & 31;
        const int lo   = lane & 15, hi = lane >> 4;
        const int mrow = (wv & 1) * 16;
        const int ncol = ((wv >> 1) & 1) * 16;
        const int kb0  = (wv >> 2) * 256;

        v8f acc = {0.f, 0.f, 0.f, 0.f, 0.f, 0.f, 0.f, 0.f};
        for (int kb = kb0; kb < kb0 + 256; kb += 4) {
            const int t0 = kb + 2 * hi;
            const float* pa = &sXM[t0 * 40 + mrow + lo];
            const float* pb = &sXM[t0 * 40 + ncol + lo];
            v2f a, b;
            a.x = pa[0];  a.y = pa[40];
            b.x = pb[0];  b.y = pb[40];
            acc = __builtin_amdgcn_wmma_f32_16x16x4_f32(
                false, a, false, b, (short)0, acc, false, false);
        }
#pragma unroll
        for (int v = 0; v < 8; ++v) sCSCR[wv * 256 + v * 32 + lane] = acc[v];
        __syncthreads();
        if (wv < 4) {
#pragma unroll
            for (int v = 0; v < 8; ++v) {
                float s = sCSCR[wv * 256 + v * 32 + lane]
                        + sCSCR[(wv + 4) * 256 + v * 32 + lane];
                const int M = mrow + v + 8 * hi;
                const int N = ncol + lo;
                sEMP[M * 33 + N] = s * (1.0f / 510.0f);
            }
        }
    }
    __syncthreads();

    // ---------------- shrinkage scalars ----------------
    {
        float psq = 0.f;
#pragma unroll
        for (int r = 0; r < 4; ++r) {
            int idx = tid + r * NTHR;           // 0..1023
            float v = sEMP[(idx >> 5) * 33 + (idx & 31)];
            psq += v * v;
        }
        sRed[tid] = psq;
        __syncthreads();
        for (int s = NTHR / 2; s > 0; s >>= 1) {
            if (tid < s) sRed[tid] += sRed[tid + s];
            __syncthreads();
        }
        if (tid == 0) {
            float tr = 0.f;
            for (int i = 0; i < 32; ++i) tr += sEMP[i * 33 + i];
            float mu    = tr * (1.0f / 32.0f);
            float alpha = sRed[0] * (1.0f / 1024.0f);
            float num   = alpha + mu * mu;
            float den   = 512.0f * (alpha - mu * mu * (1.0f / 32.0f));  // (T+1)=512
            float shrink = (den == 0.f) ? 1.0f : fminf(num / den, 1.0f);
            sScal[0] = shrink;
            sScal[1] = mu;
        }
        __syncthreads();
    }

    // ---------------- shrunk matrix (in place over emp) ----------------
    {
        float shrink = sScal[0], mu = sScal[1];
#pragma unroll
        for (int r = 0; r < 4; ++r) {
            int idx = tid + r * NTHR;
            int i = idx >> 5, j = idx & 31;
            float v = (1.0f - shrink) * sEMP[i * 33 + j];
            if (i == j) v += shrink * mu;
            sEMP[i * 33 + j] = v;
        }
        __syncthreads();
    }

    // ---------------- 32x32 Cholesky on wave 0 (thread i owns row i) ----------------
    if (tid < 32) {
        float a[32];
#pragma unroll
        for (int j = 0; j < 32; ++j) a[j] = sEMP[tid * 33 + j];
#pragma unroll
        for (int k = 0; k < 32; ++k) {
            float akk = __shfl(a[k], k, 32);
            float dkk = sqrtf(akk);
            float inv = 1.0f / dkk;
            float v   = a[k] * inv;
            a[k] = (tid == k) ? dkk : v;
#pragma unroll
            for (int j = k + 1; j < 32; ++j) {
                float ljk = __shfl(a[k], j, 32);
                a[j] = fmaf(-a[k], ljk, a[j]);
            }
        }
#pragma unroll
        for (int j = 0; j < 32; ++j)
            if (j <= tid) sEMP[tid * 33 + j] = a[j];
    }
    __syncthreads();

    // ---------------- write outputs ----------------
    if (tid < 32)
        outD[(size_t)row * 32 + tid] = sEMP[tid * 33 + tid];
    for (int p = tid; p < 496; p += NTHR) {
        float fi = (sqrtf(8.0f * (float)p + 1.0f) + 1.0f) * 0.5f;
        int i = (int)fi;
        while (i * (i - 1) / 2 > p) --i;
        while (i * (i + 1) / 2 <= p) ++i;
        int j = p - i * (i - 1) / 2;
        outL[(size_t)row * 496 + p] = sEMP[i * 33 + j];
    }
}

extern "C" void kernel_launch(void* const* d_in, const int* in_sizes, int n_in,
                              void* d_out, int out_size, void* d_ws, size_t ws_size,
                              hipStream_t stream) {
    (void)n_in; (void)out_size; (void)d_ws; (void)ws_size;
    const float* x   = (const float*)d_in[0];
    const float* w0  = (const float*)d_in[1];
    const float* b0  = (const float*)d_in[2];
    const float* g0  = (const float*)d_in[3];
    const float* be0 = (const float*)d_in[4];
    const float* w1  = (const float*)d_in[5];
    const float* b1  = (const float*)d_in[6];
    const float* g1  = (const float*)d_in[7];
    const float* be1 = (const float*)d_in[8];
    const float* w2  = (const float*)d_in[9];
    const float* b2  = (const float*)d_in[10];
    const float* g2  = (const float*)d_in[11];
    const float* be2 = (const float*)d_in[12];
    const float* w3  = (const float*)d_in[13];
    const float* b3  = (const float*)d_in[14];

    const int rows = in_sizes[0] / (9 * 512);   // 2048
    float* outD = (float*)d_out;
    float* outL = outD + (size_t)rows * 32;

    encoder_fused_kernel<<<dim3(rows), dim3(NTHR), 0, stream>>>(
        x, w0, b0, g0, be0, w1, b1, g1, be1, w2, b2, g2, be2, w3, b3,
        outD, outL);
}